// DiscriminativeLossWHardNegatives_59279138619706
// MI455X (gfx1250) — compile-verified
//
#include <hip/hip_runtime.h>

typedef __attribute__((ext_vector_type(2))) float v2f;
typedef __attribute__((ext_vector_type(8))) float v8f;

#define TPB 256

__global__ __launch_bounds__(TPB) void disc_loss_hardneg_kernel(
    const float* __restrict__ recv,    // (B, E) f32
    const float* __restrict__ emb,     // (V, E) f32
    const int*   __restrict__ nns,     // (V, Knns) int32
    const int*   __restrict__ labels,  // (B,) int32
    const int*   __restrict__ nh_ptr,  // scalar: num_hard_negatives
    float* __restrict__ out_loss,      // (B,)
    float* __restrict__ out_acc,       // (B,)
    int E, int Knns)
{
    extern __shared__ float smem[];
    float* ro_lds = smem;           // E floats (normalized receiver row)
    float* logits = ro_lds + E;     // 128 floats
    float* red    = logits + 128;   // 512 floats (two reduce banks of 256)
    float* stats  = red + 512;      // small scratch

    const int tid = threadIdx.x;
    const int b   = blockIdx.x;

    int Ksel = nh_ptr[0] + 1;
    if (Ksel > 128) Ksel = 128;
    const int numTiles = (Ksel + 15) >> 4;

    // ---- Phase 1: load ro row, block-reduce sum-of-squares and max ----
    const float* rrow = recv + (size_t)b * E;
    float ss = 0.0f;
    float mx = -__builtin_inff();
    for (int i = tid; i < E; i += TPB) {
        float v = rrow[i];
        ro_lds[i] = v;
        ss += v * v;
        mx = fmaxf(mx, v);
    }
    red[tid]       = ss;
    red[TPB + tid] = mx;
    __syncthreads();
    for (int s = TPB / 2; s > 0; s >>= 1) {
        if (tid < s) {
            red[tid]       = red[tid] + red[tid + s];
            red[TPB + tid] = fmaxf(red[TPB + tid], red[TPB + tid + s]);
        }
        __syncthreads();
    }
    if (tid == 0) {
        stats[0] = 1.0f / sqrtf(red[0]);                 // 1/||ro||
        // argmax(ro)==0  <=>  ro[0] equals the global max (index 0 wins ties)
        out_acc[b] = (ro_lds[0] >= red[TPB]) ? 1.0f : 0.0f;
    }
    __syncthreads();
    const float inv = stats[0];
    for (int i = tid; i < E; i += TPB) ro_lds[i] *= inv;
    __syncthreads();

    // ---- Phase 2: WMMA logits. Wave w owns neighbor tile [16w, 16w+16) ----
    const int wave = tid >> 5;
    const int lane = tid & 31;
    const int row  = lane & 15;             // M row within tile
    const int koff = (lane >> 4) << 1;      // K offset 0 (lanes 0-15) / 2 (lanes 16-31)

    const int label = labels[b];
    const long long nbase = (long long)label * (long long)Knns;

    if (wave < numTiles) {
        const int m = wave * 16 + row;
        const int idx = (m < Ksel) ? nns[nbase + m] : nns[nbase];
        const float* arow = emb + (size_t)idx * (size_t)E + koff;
        const float* brow = ro_lds + koff;

        v8f acc = {};
        // Outer chunks of 128 elements: prefetch 512B ahead, inner fully unrolled.
        for (int e0 = 0; e0 < E; e0 += 128) {
            __builtin_prefetch(arow + e0 + 128, 0, 1);
#pragma unroll
            for (int k = 0; k < 128; k += 4) {
                v2f a  = *(const v2f*)(arow + e0 + k);   // A[16x4] frag: K=koff,koff+1
                v2f bb = *(const v2f*)(brow + e0 + k);   // B[4x16] frag: ro replicated over N
                acc = __builtin_amdgcn_wmma_f32_16x16x4_f32(
                    /*neg_a=*/false, a, /*neg_b=*/false, bb,
                    /*c_mod=*/(short)0, acc, /*reuse_a=*/false, /*reuse_b=*/false);
            }
        }
        // D layout: lane 0 VGPR r -> (M=r, N=0); lane 16 VGPR r -> (M=r+8, N=0).
        if (lane == 0) {
#pragma unroll
            for (int r = 0; r < 8; ++r) logits[wave * 16 + r] = acc[r];
        } else if (lane == 16) {
#pragma unroll
            for (int r = 0; r < 8; ++r) logits[wave * 16 + 8 + r] = acc[r];
        }
    }
    __syncthreads();

    // ---- Phase 3: log-softmax CE at index 0 ----
    const float lv = (tid < Ksel) ? logits[tid] : -__builtin_inff();
    red[tid] = lv;
    __syncthreads();
    for (int s = TPB / 2; s > 0; s >>= 1) {
        if (tid < s) red[tid] = fmaxf(red[tid], red[tid + s]);
        __syncthreads();
    }
    const float gmax = red[0];
    __syncthreads();
    red[tid] = (tid < Ksel) ? __expf(lv - gmax) : 0.0f;
    __syncthreads();
    for (int s = TPB / 2; s > 0; s >>= 1) {
        if (tid < s) red[tid] = red[tid] + red[tid + s];
        __syncthreads();
    }
    if (tid == 0) {
        // loss = logsumexp(logits) - logits[0]
        out_loss[b] = __logf(red[0]) + gmax - logits[0];
    }
}

extern "C" void kernel_launch(void* const* d_in, const int* in_sizes, int n_in,
                              void* d_out, int out_size, void* d_ws, size_t ws_size,
                              hipStream_t stream) {
    const float* recv   = (const float*)d_in[0];   // (B,E)
    const float* emb    = (const float*)d_in[1];   // (V,E)
    const int*   nns    = (const int*)d_in[2];     // (V,Knns)
    const int*   labels = (const int*)d_in[3];     // (B,)
    const int*   nh     = (const int*)d_in[4];     // scalar

    const int B    = in_sizes[3];
    const int E    = in_sizes[0] / B;
    const int V    = (int)((long long)in_sizes[1] / (long long)E);
    const int Knns = (int)((long long)in_sizes[2] / (long long)V);

    float* out_loss = (float*)d_out;
    float* out_acc  = out_loss + B;

    const size_t smem = (size_t)(E + 128 + 512 + 8) * sizeof(float);
    disc_loss_hardneg_kernel<<<B, TPB, smem, stream>>>(
        recv, emb, nns, labels, nh, out_loss, out_acc, E, Knns);
}